// GCNLayer_34583076668066
// MI455X (gfx1250) — compile-verified
//
#include <hip/hip_runtime.h>
#include <hip/hip_bf16.h>

#define USER_NUM 100000
#define ITEM_NUM 50000
#define DIM 128

typedef float v2f __attribute__((ext_vector_type(2)));
typedef float v8f __attribute__((ext_vector_type(8)));

// ---------------------------------------------------------------------------
// Zero the ue/ie accumulators (float4 stores, fully coalesced).
// ---------------------------------------------------------------------------
__global__ __launch_bounds__(256) void gcn_zero_kernel(float* __restrict__ p, long n4) {
    long i = (long)blockIdx.x * blockDim.x + threadIdx.x;
    if (i < n4) ((float4*)p)[i] = make_float4(0.f, 0.f, 0.f, 0.f);
}

// ---------------------------------------------------------------------------
// SpMM scatter-add: for each edge e, dst[seg_dst[e]] += vals[e] * src[seg_src[e]]
// One wave (32 lanes) per edge; float4 gather (512B/wave, coalesced) then
// 4 hardware global_atomic_add_f32 per lane (non-returning -> STOREcnt path).
// Working set is L2-resident on MI455X (192MB L2), so atomics resolve in L2.
// ---------------------------------------------------------------------------
__global__ __launch_bounds__(256) void gcn_spmm_kernel(
    const int*   __restrict__ seg_src,   // gather index per edge
    const int*   __restrict__ seg_dst,   // scatter index per edge
    const float* __restrict__ vals,
    const float* __restrict__ src,       // [*, 128]
    float*       __restrict__ dst,       // [*, 128] accumulated
    int n_edges)
{
    int  lane = threadIdx.x & 31;
    long e    = ((long)blockIdx.x * blockDim.x + threadIdx.x) >> 5;
    if (e >= n_edges) return;

    int   si = seg_src[e];
    int   di = seg_dst[e];
    float v  = vals[e];

    float4 m = ((const float4*)(src + (size_t)si * DIM))[lane];  // 128 floats / wave
    float* d = dst + (size_t)di * DIM + lane * 4;
    unsafeAtomicAdd(d + 0, v * m.x);
    unsafeAtomicAdd(d + 1, v * m.y);
    unsafeAtomicAdd(d + 2, v * m.z);
    unsafeAtomicAdd(d + 3, v * m.w);
}

// ---------------------------------------------------------------------------
// out[M,128] = sigmoid(A[M,128] @ W[128,128]) using V_WMMA_F32_16X16X4_F32.
// Block = 256 threads = 8 waves. Block owns a 16-row A tile (contiguous 8KB,
// staged once into LDS, reused by all 8 waves); wave w owns N tile [16w,16w+16).
// 32 K-steps of the f32 16x16x4 WMMA accumulate the 16x16 tile in v8f.
// M is a multiple of 16 (100000, 50000) so EXEC is all-ones for every WMMA.
// Epilogue sigmoid uses v_exp_f32 + v_rcp_f32 (fast TRANS path, ~1 ulp).
// ---------------------------------------------------------------------------
__global__ __launch_bounds__(256) void gcn_gemm_sigmoid_kernel(
    const float* __restrict__ A,   // [M, 128]
    const float* __restrict__ W,   // [128, 128] row-major (K x N)
    float*       __restrict__ out) // [M, 128]
{
    __shared__ float As[16 * DIM];             // 8KB A tile shared by 8 waves
    const int m0 = blockIdx.x * 16;

    // Cooperative stage: the 16x128 tile is 2048 contiguous floats of A.
    {
        const float4* g = (const float4*)(A + (size_t)m0 * DIM);
        float4*       s = (float4*)As;
        for (int i = threadIdx.x; i < 16 * DIM / 4; i += blockDim.x)
            s[i] = g[i];
    }
    __syncthreads();

    const int wave = threadIdx.x >> 5;         // 0..7 -> N tile
    const int lane = threadIdx.x & 31;
    const int half = lane >> 4;                // K sub-phase (ISA A/B layout)
    const int idx  = lane & 15;                // A row (frag A) / N col (frag B/C)
    const int n0   = wave * 16;

    const float* a_row = As + idx * DIM;
    v8f acc = {};

    #pragma unroll 8
    for (int k = 0; k < DIM; k += 4) {
        const int kk = k + half * 2;
        // A 16x4 f32: lanes 0-15 -> K=k+0/k+1, lanes 16-31 -> K=k+2/k+3
        v2f a = { a_row[kk], a_row[kk + 1] };
        // B 4x16 f32: same K split; lane%16 = N column
        v2f b = { W[(size_t)kk * DIM + n0 + idx],
                  W[(size_t)(kk + 1) * DIM + n0 + idx] };
        acc = __builtin_amdgcn_wmma_f32_16x16x4_f32(
            /*neg_a=*/false, a, /*neg_b=*/false, b,
            /*c_mod=*/(short)0, acc, /*reuse_a=*/false, /*reuse_b=*/false);
    }

    // C/D layout: VGPR r -> row M = half*8 + r, col N = idx
    float* o = out + (size_t)m0 * DIM + n0 + idx;
    #pragma unroll
    for (int r = 0; r < 8; ++r) {
        const int mr = half * 8 + r;
        float x = acc[r];
        // sigmoid(x) = rcp(1 + exp(-x)); v_exp_f32 + v_rcp_f32, no IEEE div
        float s = __builtin_amdgcn_rcpf(1.0f + __expf(-x));
        o[(size_t)mr * DIM] = s;
    }
}

// ---------------------------------------------------------------------------
extern "C" void kernel_launch(void* const* d_in, const int* in_sizes, int n_in,
                              void* d_out, int out_size, void* d_ws, size_t ws_size,
                              hipStream_t stream)
{
    (void)n_in; (void)out_size; (void)ws_size;
    const float* user_emb = (const float*)d_in[0];  (void)user_emb; // unused by ref
    const float* item_emb = (const float*)d_in[1];
    const int*   rows     = (const int*)d_in[2];
    const int*   cols     = (const int*)d_in[3];
    const float* vals     = (const float*)d_in[4];
    const float* u_w      = (const float*)d_in[5];
    const float* i_w      = (const float*)d_in[6];
    const int    n_edges  = in_sizes[2];

    // Workspace: ue [USER_NUM,128] then ie [ITEM_NUM,128]  (76.8 MB total)
    float* ue = (float*)d_ws;
    float* ie = ue + (size_t)USER_NUM * DIM;

    float* user_out = (float*)d_out;
    float* item_out = user_out + (size_t)USER_NUM * DIM;

    // 1) zero accumulators (must re-zero every call: harness doesn't re-poison)
    const long zero4 = (long)(USER_NUM + ITEM_NUM) * DIM / 4;
    gcn_zero_kernel<<<(int)((zero4 + 255) / 256), 256, 0, stream>>>(ue, zero4);

    // 2) ue = A * item_emb   (gather cols, scatter rows)
    const long spmm_threads = (long)n_edges * 32;
    const int  spmm_blocks  = (int)((spmm_threads + 255) / 256);
    gcn_spmm_kernel<<<spmm_blocks, 256, 0, stream>>>(cols, rows, vals, item_emb, ue, n_edges);

    // 3) ie = A^T * ue       (gather rows, scatter cols)
    gcn_spmm_kernel<<<spmm_blocks, 256, 0, stream>>>(rows, cols, vals, ue, ie, n_edges);

    // 4/5) dense heads: sigmoid(X @ W) via f32 WMMA
    gcn_gemm_sigmoid_kernel<<<USER_NUM / 16, 256, 0, stream>>>(ue, u_w, user_out);
    gcn_gemm_sigmoid_kernel<<<ITEM_NUM / 16, 256, 0, stream>>>(ie, i_w, item_out);
}